// PlaneNet_4853313045169
// MI455X (gfx1250) — compile-verified
//
#include <hip/hip_runtime.h>

// ---------------------------------------------------------------------------
// PlaneNet message passing for MI455X (gfx1250, wave32).
//
// Per plane: edge gate + scatter-add (L2/atomic bound), then two small GEMMs
// done with V_WMMA_F32_16X16X4_F32 (fp32-exact tensor path on CDNA5).
// ---------------------------------------------------------------------------

#define N_NODES 50000
#define N_EDGES 800000
#define D       64          // feature width
#define NTILES  (N_NODES / 16)   // 3125 node tiles of 16 rows
#define H_STRIDE 66         // padded LDS row stride (floats) -> no bank conflicts

typedef float v2f __attribute__((ext_vector_type(2)));
typedef float v8f __attribute__((ext_vector_type(8)));

// -------------------------------- zero fill --------------------------------
__global__ void zero_kernel(float4* __restrict__ p, int n4) {
    int i = blockIdx.x * blockDim.x + threadIdx.x;
    int stride = gridDim.x * blockDim.x;
    float4 z = make_float4(0.f, 0.f, 0.f, 0.f);
    for (; i < n4; i += stride) p[i] = z;
}

// ------------------------------- edge phase --------------------------------
// One wave32 per edge. Lane L owns features {2L, 2L+1}:
//   - coalesced 256B row reads of x[dst], x[src]
//   - gate dot-product via 5-step shfl_xor wave reduction
//   - 2 global f32 atomics per lane into aggr[dst]
__global__ __launch_bounds__(256) void edge_kernel(
    const float* __restrict__ x, const int* __restrict__ ei,
    const float* __restrict__ We, const float* __restrict__ be,
    float* __restrict__ aggr) {
    const int lane = threadIdx.x & 31;
    const int f    = lane * 2;
    const int wave = blockIdx.x * (blockDim.x >> 5) + (threadIdx.x >> 5);
    const int nWaves = gridDim.x * (blockDim.x >> 5);

    // We = [2D,1]: first D rows weight x_i, next D rows weight x_j
    const float2 wei = *(const float2*)(We + f);
    const float2 wej = *(const float2*)(We + D + f);
    const float  beb = be[0];

    for (int e = wave; e < N_EDGES; e += nWaves) {
        const int src = ei[e];            // ei[0][e]
        const int dst = ei[N_EDGES + e];  // ei[1][e]
        const float2 xi = *(const float2*)(x + (size_t)dst * D + f);
        const float2 xj = *(const float2*)(x + (size_t)src * D + f);
        float p = xi.x * wei.x + xi.y * wei.y + xj.x * wej.x + xj.y * wej.y;
        #pragma unroll
        for (int off = 16; off > 0; off >>= 1)
            p += __shfl_xor(p, off, 32);
        const float g = 1.f / (1.f + __expf(-(p + beb)));
        atomicAdd(aggr + (size_t)dst * D + f,     g * xj.x);
        atomicAdd(aggr + (size_t)dst * D + f + 1, g * xj.y);
    }
}

// ------------------------------- node phase --------------------------------
// 256 threads = 8 waves. Wave w: node sub-tile (w>>2), feature tile (w&3).
// GEMM1: cat(x,aggr)[16x128] @ W1[128x64]  -> h (tanh, via LDS exchange)
// GEMM2: h[16x64]            @ W2[64x64]   -> out (tanh)
// Both GEMMs chained through V_WMMA_F32_16X16X4_F32 accumulators.
__global__ __launch_bounds__(256) void node_kernel(
    const float* __restrict__ x,  const float* __restrict__ aggr,
    const float* __restrict__ W1, const float* __restrict__ b1,
    const float* __restrict__ W2, const float* __restrict__ b2,
    float* __restrict__ out) {
    __shared__ float h_lds[2 * 16 * H_STRIDE];

    const int lane    = threadIdx.x & 31;
    const int wave    = threadIdx.x >> 5;
    const int tileSel = wave >> 2;        // which 16-node sub-tile of the block
    const int ftile   = wave & 3;         // 16-col feature tile
    const int halfSel = lane >> 4;        // 0: lanes 0-15 (K0,K1), 1: lanes 16-31 (K2,K3)
    const int lid     = lane & 15;        // row within A-tile / col within C-tile
    const int col     = ftile * 16 + lid; // output feature column

    // Preload B fragments (ISA 16x16x4 layout: VGPR0/1 = rows 4k+2h, 4k+2h+1 at col)
    v2f b1f[32], b2f[16];
    #pragma unroll
    for (int k = 0; k < 32; ++k) {
        const int r = 4 * k + 2 * halfSel;
        b1f[k][0] = W1[(size_t)(r    ) * D + col];
        b1f[k][1] = W1[(size_t)(r + 1) * D + col];
    }
    #pragma unroll
    for (int k = 0; k < 16; ++k) {
        const int r = 4 * k + 2 * halfSel;
        b2f[k][0] = W2[(size_t)(r    ) * D + col];
        b2f[k][1] = W2[(size_t)(r + 1) * D + col];
    }
    const float b1c = b1[col], b2c = b2[col];

    const int nPairs = (NTILES + 1) / 2;
    for (int pair = blockIdx.x; pair < nPairs; pair += gridDim.x) {
        const int tile  = pair * 2 + tileSel;
        const bool valid = (tile < NTILES);
        const int nodeBase = tile * 16;
        float* hbase = h_lds + tileSel * (16 * H_STRIDE);

        if (valid) {
            // ---- GEMM1: A = [x | aggr], K = 128 ----
            v8f c1 = {0.f, 0.f, 0.f, 0.f, 0.f, 0.f, 0.f, 0.f};
            #pragma unroll
            for (int k = 0; k < 32; ++k) {
                const float* mat = (k < 16) ? x : aggr;     // first/second half of cat()
                const int kk = (k & 15) * 4 + 2 * halfSel;  // K offset within half
                const float2 av = *(const float2*)(mat + (size_t)(nodeBase + lid) * D + kk);
                v2f a; a[0] = av.x; a[1] = av.y;
                c1 = __builtin_amdgcn_wmma_f32_16x16x4_f32(
                        false, a, false, b1f[k], (short)0, c1, false, false);
            }
            // bias + tanh, stash h tile to LDS (C/D layout: VGPR r -> rows r, r+8)
            #pragma unroll
            for (int r = 0; r < 8; ++r)
                hbase[(r + 8 * halfSel) * H_STRIDE + col] = tanhf(c1[r] + b1c);
        }
        __syncthreads();

        if (valid) {
            // ---- GEMM2: A = h, K = 64 ----
            v8f c2 = {0.f, 0.f, 0.f, 0.f, 0.f, 0.f, 0.f, 0.f};
            #pragma unroll
            for (int k = 0; k < 16; ++k) {
                v2f a = *(const v2f*)(hbase + lid * H_STRIDE + 4 * k + 2 * halfSel);
                c2 = __builtin_amdgcn_wmma_f32_16x16x4_f32(
                        false, a, false, b2f[k], (short)0, c2, false, false);
            }
            #pragma unroll
            for (int r = 0; r < 8; ++r) {
                const int m = r + 8 * halfSel;
                out[(size_t)(nodeBase + m) * D + col] = tanhf(c2[r] + b2c);
            }
        }
        __syncthreads();  // protect h_lds before next grid-stride iteration
    }
}

// ------------------------------ host launcher ------------------------------
// setup_inputs() dict order: per plane p in {u,v,y}:
//   [8p+0]=x  [8p+1]=ei(int32)  [8p+2]=We  [8p+3]=be
//   [8p+4]=W1 [8p+5]=b1         [8p+6]=W2  [8p+7]=b2
// d_out = [3, N, D] f32.  d_ws: one reusable N*D f32 aggr buffer (12.8 MB).
extern "C" void kernel_launch(void* const* d_in, const int* in_sizes, int n_in,
                              void* d_out, int out_size, void* d_ws, size_t ws_size,
                              hipStream_t stream) {
    (void)in_sizes; (void)n_in; (void)out_size; (void)ws_size;
    float* aggr = (float*)d_ws;
    const int nPairs = (NTILES + 1) / 2;  // 1563 blocks covers all node tiles

    for (int p = 0; p < 3; ++p) {
        const float* x  = (const float*)d_in[8 * p + 0];
        const int*   ei = (const int*)  d_in[8 * p + 1];
        const float* We = (const float*)d_in[8 * p + 2];
        const float* be = (const float*)d_in[8 * p + 3];
        const float* W1 = (const float*)d_in[8 * p + 4];
        const float* b1 = (const float*)d_in[8 * p + 5];
        const float* W2 = (const float*)d_in[8 * p + 6];
        const float* b2 = (const float*)d_in[8 * p + 7];
        float* out = (float*)d_out + (size_t)p * N_NODES * D;

        zero_kernel<<<1024, 256, 0, stream>>>((float4*)aggr, N_NODES * D / 4);
        edge_kernel<<<6144, 256, 0, stream>>>(x, ei, We, be, aggr);
        node_kernel<<<nPairs, 256, 0, stream>>>(x, aggr, W1, b1, W2, b2, out);
    }
}